// GNN_1176821039615
// MI455X (gfx1250) — compile-verified
//
#include <hip/hip_runtime.h>
#include <hip/hip_bf16.h>

typedef __attribute__((ext_vector_type(2))) float v2f;
typedef __attribute__((ext_vector_type(8))) float v8f;

// ---------------------------------------------------------------------------
// GEMM: H[n,64] = X[n,64] @ W[64,64] via V_WMMA_F32_16X16X4_F32.
// Block = 128 threads = 4 waves; each wave computes a 16-row x 64-col strip
// as 4 accumulator tiles of 16x16. K=64 processed in 16 steps of K=4.
// W is staged in LDS K-pair-interleaved: Ws[(k>>1)*128 + col*2 + (k&1)],
// so each B fragment (W[k][col], W[k+1][col]) is one aligned ds_load_b64.
// ---------------------------------------------------------------------------
__global__ __launch_bounds__(128) void gcn_gemm64_wmma(
    const float* __restrict__ X, const float* __restrict__ W,
    float* __restrict__ H, int n)
{
    __shared__ __align__(16) float Ws[64 * 64];
    // cooperative load + K-pair interleave (coalesced global reads)
    for (int idx = threadIdx.x; idx < 64 * 64; idx += 128) {
        int k = idx >> 6;
        int c = idx & 63;
        Ws[((k >> 1) * 128) + (c << 1) + (k & 1)] = W[idx];
    }
    __syncthreads();

    const int wave = threadIdx.x >> 5;
    const int lane = threadIdx.x & 31;
    const int tile = blockIdx.x * 4 + wave;   // 16-row tile index
    const int row0 = tile * 16;
    if (row0 >= n) return;

    const int mrow = lane & 15;               // A: M index
    const int koff = (lane >> 4) << 1;        // A/B: K sub-offset (0 or 2)
    const int col  = lane & 15;               // B/C/D: N index
    const float* xrow = X + (size_t)(row0 + mrow) * 64;

    v8f acc0 = {}, acc1 = {}, acc2 = {}, acc3 = {};
    #pragma unroll
    for (int k = 0; k < 64; k += 4) {
        v2f a;
        a.x = xrow[k + koff];
        a.y = xrow[k + koff + 1];
        // (k + koff) is even: fragment pair is contiguous in interleaved LDS
        const v2f* wp = (const v2f*)(Ws + (((k + koff) >> 1) * 128) + (col << 1));
        v2f b0 = wp[0];
        v2f b1 = wp[16];
        v2f b2 = wp[32];
        v2f b3 = wp[48];
        acc0 = __builtin_amdgcn_wmma_f32_16x16x4_f32(false, a, false, b0, (short)0, acc0, false, false);
        acc1 = __builtin_amdgcn_wmma_f32_16x16x4_f32(false, a, false, b1, (short)0, acc1, false, false);
        acc2 = __builtin_amdgcn_wmma_f32_16x16x4_f32(false, a, false, b2, (short)0, acc2, false, false);
        acc3 = __builtin_amdgcn_wmma_f32_16x16x4_f32(false, a, false, b3, (short)0, acc3, false, false);
    }

    // C/D layout: VGPR r holds row M=r (lanes 0-15) / M=r+8 (lanes 16-31)
    float* out = H + (size_t)(row0 + (lane >> 4) * 8) * 64 + col;
    #pragma unroll
    for (int r = 0; r < 8; ++r) {
        out[(size_t)r * 64 +  0] = acc0[r];
        out[(size_t)r * 64 + 16] = acc1[r];
        out[(size_t)r * 64 + 32] = acc2[r];
        out[(size_t)r * 64 + 48] = acc3[r];
    }
}

// ---------------------------------------------------------------------------
// Degree / normalization precompute
// ---------------------------------------------------------------------------
__global__ void init_deg_kernel(float* deg, int n) {
    int i = blockIdx.x * blockDim.x + threadIdx.x;
    if (i < n) deg[i] = 1.0f;                 // self-loop weight
}

__global__ void deg_accum_kernel(const int* __restrict__ dst,
                                 const float* __restrict__ ew,
                                 float* __restrict__ deg, int E) {
    int e = blockIdx.x * blockDim.x + threadIdx.x;
    if (e < E) atomicAdd(&deg[dst[e]], ew[e]);
}

__global__ void dinv_kernel(const float* __restrict__ deg,
                            float* __restrict__ dinv, int n) {
    int i = blockIdx.x * blockDim.x + threadIdx.x;
    if (i < n) {
        float d = deg[i];
        dinv[i] = (d > 0.0f) ? rsqrtf(fmaxf(d, 1e-30f)) : 0.0f;
    }
}

__global__ void edge_norm_kernel(const int* __restrict__ src,
                                 const int* __restrict__ dst,
                                 const float* __restrict__ ew,
                                 const float* __restrict__ dinv,
                                 float* __restrict__ nrm, int E) {
    int e = blockIdx.x * blockDim.x + threadIdx.x;
    if (e < E) nrm[e] = dinv[src[e]] * ew[e] * dinv[dst[e]];
}

// ---------------------------------------------------------------------------
// Aggregation
// ---------------------------------------------------------------------------
__global__ void zero_kernel(float4* __restrict__ p, int n4) {
    int i = blockIdx.x * blockDim.x + threadIdx.x;
    if (i < n4) p[i] = make_float4(0.f, 0.f, 0.f, 0.f);
}

// One edge per wave; lane l handles features 2l, 2l+1 (coalesced float2).
__global__ __launch_bounds__(256) void scatter_kernel(
    const float* __restrict__ H, const int* __restrict__ src,
    const int* __restrict__ dst, const float* __restrict__ nrm,
    float* __restrict__ agg, int E)
{
    long long gid = (long long)blockIdx.x * 256 + threadIdx.x;
    int e = (int)(gid >> 5);
    int lane = (int)(gid & 31);
    if (e >= E) return;
    float nv = nrm[e];
    int s = src[e], d = dst[e];
    float2 v = *((const float2*)(H + (size_t)s * 64) + lane);
    float* ap = agg + (size_t)d * 64 + lane * 2;
    atomicAdd(ap + 0, v.x * nv);
    atomicAdd(ap + 1, v.y * nv);
}

// agg = relu(agg + h*dinv^2 + bias)   (self-loop folded in, no atomics)
__global__ void finalize_kernel(float* __restrict__ agg,
                                const float* __restrict__ H,
                                const float* __restrict__ dinv,
                                const float* __restrict__ bias, int total)
{
    int gid = blockIdx.x * blockDim.x + threadIdx.x;
    if (gid >= total) return;
    int node = gid >> 6;
    int f = gid & 63;
    float di = dinv[node];
    float v = agg[gid] + H[gid] * di * di + bias[f];
    agg[gid] = fmaxf(v, 0.0f);
}

// ---------------------------------------------------------------------------
// MLP head: y = relu(h @ Wm1 + bm1) @ Wm2 + bm2   (one thread per node)
// ---------------------------------------------------------------------------
__global__ __launch_bounds__(256) void mlp_kernel(
    const float* __restrict__ H, const float* __restrict__ Wm1,
    const float* __restrict__ bm1, const float* __restrict__ Wm2,
    const float* __restrict__ bm2, float* __restrict__ Y, int n)
{
    __shared__ float w1[64 * 16];
    __shared__ float b1s[16], w2s[16];
    __shared__ float b2s;
    for (int i = threadIdx.x; i < 64 * 16; i += 256) w1[i] = Wm1[i];
    if (threadIdx.x < 16) { b1s[threadIdx.x] = bm1[threadIdx.x]; w2s[threadIdx.x] = Wm2[threadIdx.x]; }
    if (threadIdx.x == 0) b2s = bm2[0];
    __syncthreads();

    int node = blockIdx.x * 256 + threadIdx.x;
    if (node >= n) return;
    const float* row = H + (size_t)node * 64;
    float t[16];
    #pragma unroll
    for (int j = 0; j < 16; ++j) t[j] = b1s[j];
    for (int k = 0; k < 64; ++k) {
        float xv = row[k];
        const float* wr = w1 + k * 16;
        #pragma unroll
        for (int j = 0; j < 16; ++j) t[j] = fmaf(xv, wr[j], t[j]);
    }
    float y = b2s;
    #pragma unroll
    for (int j = 0; j < 16; ++j) y = fmaf(fmaxf(t[j], 0.0f), w2s[j], y);
    Y[node] = y;
}

// ---------------------------------------------------------------------------
// Global min/max + normalize (order-preserving uint encoding for atomics)
// ---------------------------------------------------------------------------
__device__ inline unsigned enc_f32(float f) {
    unsigned u = __float_as_uint(f);
    return (u >> 31) ? ~u : (u | 0x80000000u);
}
__device__ inline float dec_f32(unsigned u) {
    return (u >> 31) ? __uint_as_float(u ^ 0x80000000u) : __uint_as_float(~u);
}

__global__ void minmax_init_kernel(unsigned* mm) {
    if (threadIdx.x == 0) { mm[0] = 0xFFFFFFFFu; mm[1] = 0u; }
}

__global__ __launch_bounds__(256) void minmax_reduce_kernel(
    const float* __restrict__ Y, int n, unsigned* __restrict__ mm)
{
    __shared__ unsigned smin[256], smax[256];
    unsigned mn = 0xFFFFFFFFu, mx = 0u;
    for (int i = blockIdx.x * 256 + threadIdx.x; i < n; i += gridDim.x * 256) {
        unsigned e = enc_f32(Y[i]);
        mn = min(mn, e); mx = max(mx, e);
    }
    smin[threadIdx.x] = mn; smax[threadIdx.x] = mx;
    __syncthreads();
    for (int s = 128; s > 0; s >>= 1) {
        if (threadIdx.x < s) {
            smin[threadIdx.x] = min(smin[threadIdx.x], smin[threadIdx.x + s]);
            smax[threadIdx.x] = max(smax[threadIdx.x], smax[threadIdx.x + s]);
        }
        __syncthreads();
    }
    if (threadIdx.x == 0) {
        atomicMin(&mm[0], smin[0]);
        atomicMax(&mm[1], smax[0]);
    }
}

__global__ void normalize_kernel(float* __restrict__ Y, int n,
                                 const unsigned* __restrict__ mm)
{
    int i = blockIdx.x * blockDim.x + threadIdx.x;
    if (i < n) {
        float mn = dec_f32(mm[0]);
        float mx = dec_f32(mm[1]);
        Y[i] = (Y[i] - mn) / (mx - mn);
    }
}

// ---------------------------------------------------------------------------
// Launcher
// ---------------------------------------------------------------------------
extern "C" void kernel_launch(void* const* d_in, const int* in_sizes, int n_in,
                              void* d_out, int out_size, void* d_ws, size_t ws_size,
                              hipStream_t stream)
{
    const float* x   = (const float*)d_in[0];
    const int*   ei  = (const int*)d_in[1];      // [2, E] row-major
    const float* ew  = (const float*)d_in[2];
    const float* W1  = (const float*)d_in[3];
    const float* b1  = (const float*)d_in[4];
    const float* W2  = (const float*)d_in[5];
    const float* b2  = (const float*)d_in[6];
    const float* W3  = (const float*)d_in[7];
    const float* b3  = (const float*)d_in[8];
    const float* Wm1 = (const float*)d_in[9];
    const float* bm1 = (const float*)d_in[10];
    const float* Wm2 = (const float*)d_in[11];
    const float* bm2 = (const float*)d_in[12];

    const int n = in_sizes[0] / 64;   // N nodes
    const int E = in_sizes[2];        // edges
    const int* srcI = ei;
    const int* dstI = ei + E;

    // Workspace carve-out (all offsets 16B aligned for the sizes involved)
    char* ws = (char*)d_ws;
    float* deg  = (float*)ws;  ws += (size_t)n * sizeof(float);
    float* dinv = (float*)ws;  ws += (size_t)n * sizeof(float);
    float* nrm  = (float*)ws;  ws += (size_t)E * sizeof(float);
    float* hbuf = (float*)ws;  ws += (size_t)n * 64 * sizeof(float);
    float* agg  = (float*)ws;  ws += (size_t)n * 64 * sizeof(float);
    unsigned* mm = (unsigned*)ws;

    const int T = 256;
    const int nb_n   = (n + T - 1) / T;
    const int nb_e   = (E + T - 1) / T;
    const int nb_nf  = (n * 64 + T - 1) / T;
    const int nb_nf4 = (n * 16 + T - 1) / T;
    const int nb_sc  = (int)(((long long)E * 32 + T - 1) / T);
    const int tiles  = (n + 15) / 16;
    const int nb_gm  = (tiles + 3) / 4;

    // Degree + symmetric normalization (computed once, reused for 3 layers)
    init_deg_kernel<<<nb_n, T, 0, stream>>>(deg, n);
    deg_accum_kernel<<<nb_e, T, 0, stream>>>(dstI, ew, deg, E);
    dinv_kernel<<<nb_n, T, 0, stream>>>(deg, dinv, n);
    edge_norm_kernel<<<nb_e, T, 0, stream>>>(srcI, dstI, ew, dinv, nrm, E);

    const float* Wl[3] = {W1, W2, W3};
    const float* bl[3] = {b1, b2, b3};
    const float* cur = x;
    for (int l = 0; l < 3; ++l) {
        gcn_gemm64_wmma<<<nb_gm, 128, 0, stream>>>(cur, Wl[l], hbuf, n);
        zero_kernel<<<nb_nf4, T, 0, stream>>>((float4*)agg, n * 16);
        scatter_kernel<<<nb_sc, T, 0, stream>>>(hbuf, srcI, dstI, nrm, agg, E);
        finalize_kernel<<<nb_nf, T, 0, stream>>>(agg, hbuf, dinv, bl[l], n * 64);
        cur = agg;
    }

    float* y = (float*)d_out;
    mlp_kernel<<<nb_n, T, 0, stream>>>(agg, Wm1, bm1, Wm2, bm2, y, n);
    minmax_init_kernel<<<1, 32, 0, stream>>>(mm);
    minmax_reduce_kernel<<<256, T, 0, stream>>>(y, n, mm);
    normalize_kernel<<<nb_n, T, 0, stream>>>(y, n, mm);
}